// sage_70789650972706
// MI455X (gfx1250) — compile-verified
//
#include <hip/hip_runtime.h>
#include <hip/hip_bf16.h>
#include <math.h>

typedef float v2f __attribute__((ext_vector_type(2)));
typedef float v8f __attribute__((ext_vector_type(8)));
typedef unsigned int v4u __attribute__((ext_vector_type(4)));
typedef int v4i __attribute__((ext_vector_type(4)));
typedef int v8i __attribute__((ext_vector_type(8)));

#define WMMA_F32(a, b, c) \
    __builtin_amdgcn_wmma_f32_16x16x4_f32(false, (a), false, (b), (short)0, (c), false, false)

// ---------------------------------------------------------------------------
// TDM: DMA a contiguous run of `nelem` f32 (nelem < 65536) from global memory
// into LDS at byte offset lds_off.  D# layout per CDNA5 ISA ch.8:
//   group0: count=1 | lds_addr | global_addr[56:0] | type=2
//   group1: data_size=4B, tensor_dim0=tile_dim0=nelem, tensor_dim1=tile_dim1=1
// Issued once by the executing wave; tracked with TENSORcnt.
// This toolchain's builtin takes 6 args (g0, g1, g2, g3, g4, cpol).
// ---------------------------------------------------------------------------
__device__ __forceinline__ void tdm_load_f32(const float* gptr, unsigned lds_off,
                                             unsigned nelem) {
    unsigned long long ga = (unsigned long long)(uintptr_t)gptr;
    v4u g0;
    g0.x = 1u;                                   // count=1, user descriptor
    g0.y = lds_off;                              // LDS byte address
    g0.z = (unsigned)(ga & 0xFFFFFFFFu);         // global_addr[31:0]
    g0.w = (unsigned)((ga >> 32) & 0x01FFFFFFull) | (2u << 30);  // addr[56:32]|type=2
    v8i g1;
    g1[0] = (int)(2u << 16);                     // data_size = 4 bytes
    g1[1] = (int)((nelem & 0xFFFFu) << 16);      // tensor_dim0[15:0]
    g1[2] = (int)(((nelem >> 16) & 0xFFFFu) | (1u << 16));   // dim0_hi | tensor_dim1=1
    g1[3] = (int)((nelem & 0xFFFFu) << 16);      // tensor_dim1_hi=0 | tile_dim0=nelem
    g1[4] = 1;                                   // tile_dim1=1, tile_dim2=0
    g1[5] = (int)nelem;                          // tensor_dim0_stride[31:0]
    g1[6] = 0;
    g1[7] = 0;
    v4i z4 = {0, 0, 0, 0};
    v8i z8 = {0, 0, 0, 0, 0, 0, 0, 0};
    __builtin_amdgcn_tensor_load_to_lds(g0, g1, z4, z4, z8, 0);
}

__device__ __forceinline__ unsigned lds_offset_of(const void* p) {
    // generic address of a __shared__ object: low 32 bits are the LDS offset
    return (unsigned)(uintptr_t)p;
}

// ---------------------------------------------------------------------------
// Scatter: one thread per (edge, float4 chunk): b128 gather of x[src],
// 4x global_atomic_add_f32 into agg[dst]; chunk 0 accumulates degree.
// ---------------------------------------------------------------------------
__global__ void sage_scatter64(const float* __restrict__ x,
                               const int* __restrict__ src,
                               const int* __restrict__ dst,
                               float* __restrict__ agg,
                               float* __restrict__ deg,
                               unsigned int total) {
    unsigned int idx = blockIdx.x * blockDim.x + threadIdx.x;
    if (idx >= total) return;
    unsigned int e = idx >> 4;       // 16 chunks of 4 floats = 64 features
    unsigned int c = idx & 15u;
    int s = src[e];
    int d = dst[e];
    const float4 v = *(const float4*)(x + (size_t)s * 64 + c * 4);
    float* a = agg + (size_t)d * 64 + c * 4;
    atomicAdd(a + 0, v.x);
    atomicAdd(a + 1, v.y);
    atomicAdd(a + 2, v.z);
    atomicAdd(a + 3, v.w);
    if (c == 0) atomicAdd(&deg[d], 1.0f);
}

__global__ void sage_scatter32(const float* __restrict__ h,
                               const int* __restrict__ src,
                               const int* __restrict__ dst,
                               float* __restrict__ agg,
                               unsigned int total) {
    unsigned int idx = blockIdx.x * blockDim.x + threadIdx.x;
    if (idx >= total) return;
    unsigned int e = idx >> 3;       // 8 chunks of 4 floats = 32 features
    unsigned int c = idx & 7u;
    int s = src[e];
    int d = dst[e];
    const float4 v = *(const float4*)(h + (size_t)s * 32 + c * 4);
    float* a = agg + (size_t)d * 32 + c * 4;
    atomicAdd(a + 0, v.x);
    atomicAdd(a + 1, v.y);
    atomicAdd(a + 2, v.z);
    atomicAdd(a + 3, v.w);
}

// ---------------------------------------------------------------------------
// Layer 1: h1 = relu(mean @ W_l1 + x @ W_r1 + b1)     [N,64]x[64,32] -> [N,32]
// Weights staged into LDS by the TDM; one wave per 16-node M-tile; K=64 in
// steps of 4; two 16-wide N tiles (hidden=32).
// A layout (32-bit 16x4): lane l: m=l&15, element i holds K = k0 + 2*(l>>4)+i.
// B layout (4x16): lane l: n=l&15, element i holds K = k0 + 2*(l>>4)+i.
// C/D layout: element v on lane l is (M = v + 8*(l>>4), N = l&15).
// ---------------------------------------------------------------------------
__global__ void sage_gemm_l1(const float* __restrict__ agg,
                             const float* __restrict__ deg,
                             const float* __restrict__ x,
                             const float* __restrict__ Wl,
                             const float* __restrict__ Wr,
                             const float* __restrict__ bias,
                             float* __restrict__ h1,
                             int n_nodes) {
    __shared__ float sW[2 * 64 * 32];            // [0..2047]=Wl, [2048..4095]=Wr

    if ((threadIdx.x >> 5) == 0) {               // wave 0 issues both TDM copies
        unsigned base = lds_offset_of(sW);
        tdm_load_f32(Wl, base, 64 * 32);
        tdm_load_f32(Wr, base + 64 * 32 * 4, 64 * 32);
        __builtin_amdgcn_s_wait_tensorcnt(0);
    }
    __syncthreads();

    int wave  = (int)((blockIdx.x * blockDim.x + threadIdx.x) >> 5);
    int lane  = threadIdx.x & 31;
    int node0 = wave * 16;
    if (node0 >= n_nodes) return;  // uniform per wave -> EXEC stays all-ones

    int n    = lane & 15;
    int half = lane >> 4;
    int row  = node0 + n;
    float rd = 1.0f / fmaxf(deg[row], 1.0f);

    const float* aggRow = agg + (size_t)row * 64;
    const float* xRow   = x   + (size_t)row * 64;

    v8f acc0 = {};
    v8f acc1 = {};
    #pragma unroll
    for (int k0 = 0; k0 < 64; k0 += 4) {
        int ka = k0 + half * 2;
        v2f am = *(const v2f*)(aggRow + ka);
        am.x *= rd; am.y *= rd;
        v2f ax = *(const v2f*)(xRow + ka);

        v2f bl0, bl1, br0, br1;
        bl0.x = sW[ka * 32 + n];             bl0.y = sW[(ka + 1) * 32 + n];
        bl1.x = sW[ka * 32 + 16 + n];        bl1.y = sW[(ka + 1) * 32 + 16 + n];
        br0.x = sW[2048 + ka * 32 + n];      br0.y = sW[2048 + (ka + 1) * 32 + n];
        br1.x = sW[2048 + ka * 32 + 16 + n]; br1.y = sW[2048 + (ka + 1) * 32 + 16 + n];

        acc0 = WMMA_F32(am, bl0, acc0);
        acc0 = WMMA_F32(ax, br0, acc0);
        acc1 = WMMA_F32(am, bl1, acc1);
        acc1 = WMMA_F32(ax, br1, acc1);
    }

    float b0 = bias[n];
    float b1 = bias[16 + n];
    #pragma unroll
    for (int v = 0; v < 8; ++v) {
        int M = v + 8 * half;
        float o0 = fmaxf(acc0[v] + b0, 0.0f);
        float o1 = fmaxf(acc1[v] + b1, 0.0f);
        h1[(size_t)(node0 + M) * 32 + n]      = o0;
        h1[(size_t)(node0 + M) * 32 + 16 + n] = o1;
    }
}

// ---------------------------------------------------------------------------
// Layer 2: logits = mean2 @ W_l2 + h1 @ W_r2 + b2     [N,32]x[32,40] -> [N,40]
// Three 16-wide N tiles cover cols 0..47; masked to 40 real classes.
// ---------------------------------------------------------------------------
__global__ void sage_gemm_l2(const float* __restrict__ agg,
                             const float* __restrict__ deg,
                             const float* __restrict__ h1,
                             const float* __restrict__ Wl,
                             const float* __restrict__ Wr,
                             const float* __restrict__ bias,
                             float* __restrict__ out,
                             int n_nodes) {
    __shared__ float sW[2 * 32 * 40];            // [0..1279]=Wl2, [1280..2559]=Wr2

    if ((threadIdx.x >> 5) == 0) {
        unsigned base = lds_offset_of(sW);
        tdm_load_f32(Wl, base, 32 * 40);
        tdm_load_f32(Wr, base + 32 * 40 * 4, 32 * 40);
        __builtin_amdgcn_s_wait_tensorcnt(0);
    }
    __syncthreads();

    int wave  = (int)((blockIdx.x * blockDim.x + threadIdx.x) >> 5);
    int lane  = threadIdx.x & 31;
    int node0 = wave * 16;
    if (node0 >= n_nodes) return;

    int n    = lane & 15;
    int half = lane >> 4;
    int row  = node0 + n;
    float rd = 1.0f / fmaxf(deg[row], 1.0f);

    const float* aggRow = agg + (size_t)row * 32;
    const float* hRow   = h1  + (size_t)row * 32;

    v8f acc[3];
    acc[0] = (v8f){}; acc[1] = (v8f){}; acc[2] = (v8f){};

    #pragma unroll
    for (int k0 = 0; k0 < 32; k0 += 4) {
        int ka = k0 + half * 2;
        v2f am = *(const v2f*)(aggRow + ka);
        am.x *= rd; am.y *= rd;
        v2f ah = *(const v2f*)(hRow + ka);

        #pragma unroll
        for (int t = 0; t < 3; ++t) {
            int col = t * 16 + n;
            v2f bl, br;
            bl.x = (col < 40) ? sW[ka * 40 + col]              : 0.0f;
            bl.y = (col < 40) ? sW[(ka + 1) * 40 + col]        : 0.0f;
            br.x = (col < 40) ? sW[1280 + ka * 40 + col]       : 0.0f;
            br.y = (col < 40) ? sW[1280 + (ka + 1) * 40 + col] : 0.0f;
            acc[t] = WMMA_F32(am, bl, acc[t]);
            acc[t] = WMMA_F32(ah, br, acc[t]);
        }
    }

    #pragma unroll
    for (int t = 0; t < 3; ++t) {
        int col = t * 16 + n;
        if (col >= 40) continue;
        float b = bias[col];
        #pragma unroll
        for (int v = 0; v < 8; ++v) {
            int M = v + 8 * half;
            out[(size_t)(node0 + M) * 40 + col] = acc[t][v] + b;
        }
    }
}

// ---------------------------------------------------------------------------
// In-place log_softmax over 40 classes, one wave32 per node.
// Lane handles col=lane and (lane<8) col=lane+32; shfl_xor reductions.
// ---------------------------------------------------------------------------
__global__ void sage_logsoftmax(float* __restrict__ out, int n_nodes) {
    int wave = (int)((blockIdx.x * blockDim.x + threadIdx.x) >> 5);
    int lane = threadIdx.x & 31;
    if (wave >= n_nodes) return;

    float* rowp = out + (size_t)wave * 40;
    float v0 = rowp[lane];
    float v1 = (lane < 8) ? rowp[32 + lane] : -3.4e38f;

    float m = fmaxf(v0, v1);
    #pragma unroll
    for (int off = 16; off > 0; off >>= 1)
        m = fmaxf(m, __shfl_xor(m, off, 32));

    float s = expf(v0 - m) + ((lane < 8) ? expf(v1 - m) : 0.0f);
    #pragma unroll
    for (int off = 16; off > 0; off >>= 1)
        s += __shfl_xor(s, off, 32);

    float lse = m + logf(s);
    rowp[lane] = v0 - lse;
    if (lane < 8) rowp[32 + lane] = v1 - lse;
}

// ---------------------------------------------------------------------------
extern "C" void kernel_launch(void* const* d_in, const int* in_sizes, int n_in,
                              void* d_out, int out_size, void* d_ws, size_t ws_size,
                              hipStream_t stream) {
    const float* x    = (const float*)d_in[0];
    const int*   ei   = (const int*)d_in[1];
    const float* W_l1 = (const float*)d_in[2];
    const float* W_r1 = (const float*)d_in[3];
    const float* b1   = (const float*)d_in[4];
    const float* W_l2 = (const float*)d_in[5];
    const float* W_r2 = (const float*)d_in[6];
    const float* b2   = (const float*)d_in[7];
    float*       out  = (float*)d_out;

    const int n_nodes = in_sizes[0] / 64;   // N_FEAT = 64
    const int n_edges = in_sizes[1] / 2;
    const int* src = ei;
    const int* dst = ei + n_edges;

    // Workspace layout (floats): agg[N*64] | deg[N] | h1[N*32]
    float* agg = (float*)d_ws;
    float* deg = agg + (size_t)n_nodes * 64;
    float* h1  = deg + (size_t)n_nodes;

    // ---- Layer 1 ----
    (void)hipMemsetAsync(d_ws, 0, (size_t)n_nodes * 65 * sizeof(float), stream);
    {
        unsigned int total = (unsigned int)n_edges * 16u;   // edge x float4-chunk
        unsigned int blocks = (total + 255u) / 256u;
        sage_scatter64<<<blocks, 256, 0, stream>>>(x, src, dst, agg, deg, total);
    }
    {
        int waves  = (n_nodes + 15) / 16;          // one wave per 16-node tile
        int blocks = (waves + 7) / 8;              // 8 waves (256 thr) per block
        sage_gemm_l1<<<blocks, 256, 0, stream>>>(agg, deg, x, W_l1, W_r1, b1,
                                                 h1, n_nodes);
    }

    // ---- Layer 2 (reuse agg's first N*32 floats) ----
    (void)hipMemsetAsync(agg, 0, (size_t)n_nodes * 32 * sizeof(float), stream);
    {
        unsigned int total = (unsigned int)n_edges * 8u;
        unsigned int blocks = (total + 255u) / 256u;
        sage_scatter32<<<blocks, 256, 0, stream>>>(h1, src, dst, agg, total);
    }
    {
        int waves  = (n_nodes + 15) / 16;
        int blocks = (waves + 7) / 8;
        sage_gemm_l2<<<blocks, 256, 0, stream>>>(agg, deg, h1, W_l2, W_r2, b2,
                                                 out, n_nodes);
    }

    // ---- log_softmax in place ----
    {
        int blocks = (n_nodes + 7) / 8;            // 8 waves per block
        sage_logsoftmax<<<blocks, 256, 0, stream>>>(out, n_nodes);
    }
}